// HeteroScorePredictor_6133213298979
// MI455X (gfx1250) — compile-verified
//
#include <hip/hip_runtime.h>
#include <hip/hip_bf16.h>

typedef __attribute__((ext_vector_type(2))) float v2f;
typedef __attribute__((ext_vector_type(4))) float v4f;
typedef __attribute__((ext_vector_type(8))) float v8f;

#define D_DIM 128
#define TILE  16
#define WAVES_PER_BLOCK 8

// One wave32 computes 16 edge dot-products of one edge type via
// V_WMMA_F32_16X16X4_F32: A = u-rows (16xK), B = v-rows (Kx16), diag(D) = scores.
// Grid: x = tile blocks, y = edge type (0..2)  -> no integer division anywhere.
__global__ __launch_bounds__(256) void hetero_udotv_wmma_kernel(
    const float* __restrict__ h_user, const float* __restrict__ h_item,
    const int* __restrict__ src0, const int* __restrict__ dst0,   // clicks: user->item
    const int* __restrict__ src1, const int* __restrict__ dst1,   // clickedby: item->user
    const int* __restrict__ src2, const int* __restrict__ dst2,   // follows: user->user
    float* __restrict__ out, int E)
{
    const int lane      = threadIdx.x & 31;
    const int waveInBlk = threadIdx.x >> 5;
    const int etype     = blockIdx.y;                       // scalar, uniform
    const int tile      = blockIdx.x * WAVES_PER_BLOCK + waveInBlk;

    const int tilesPer = (E + TILE - 1) / TILE;             // 31250 for E=500000
    if (tile >= tilesPer) return;                           // wave-uniform guard
    const int e0 = tile * TILE;

    const float* srcTab; const float* dstTab;
    const int*   srcIdx; const int*   dstIdx;
    if (etype == 0)      { srcTab = h_user; dstTab = h_item; srcIdx = src0; dstIdx = dst0; }
    else if (etype == 1) { srcTab = h_item; dstTab = h_user; srcIdx = src1; dstIdx = dst1; }
    else                 { srcTab = h_user; dstTab = h_user; srcIdx = src2; dstIdx = dst2; }

    const int r    = lane & 15;   // edge-in-tile: M row for A, N col for B
    const int half = lane >> 4;   // selects K sub-slots per ISA f32 WMMA layout

    // Clamp (keeps EXEC full for a tail tile; duplicated edges are harmless).
    int eMy = e0 + r; if (eMy > E - 1) eMy = E - 1;
    const long iu = (long)srcIdx[eMy];
    const long iv = (long)dstIdx[eMy];

    // Each lane walks its assigned row; float4 at column kk+4*half covers the
    // lane's share; .xy feeds WMMA #1's K-slots, .zw feeds WMMA #2's. The
    // K-slot -> memory-column mapping is identical for A and B, and dot
    // products are permutation-invariant over K, so diag(D) is exact.
    const float* uRow = srcTab + iu * (long)D_DIM + 4 * half;
    const float* vRow = dstTab + iv * (long)D_DIM + 4 * half;

    v8f acc0 = {};
    v8f acc1 = {};
#pragma unroll
    for (int kk = 0; kk < D_DIM; kk += 8) {
        v4f a4 = *(const v4f*)(uRow + kk);
        v4f b4 = *(const v4f*)(vRow + kk);
        v2f a0 = { a4.x, a4.y }, b0 = { b4.x, b4.y };
        v2f a1 = { a4.z, a4.w }, b1 = { b4.z, b4.w };
        // 8 args: (neg_a, A, neg_b, B, c_mod, C, reuse_a, reuse_b)
        acc0 = __builtin_amdgcn_wmma_f32_16x16x4_f32(false, a0, false, b0,
                                                     (short)0, acc0, false, false);
        acc1 = __builtin_amdgcn_wmma_f32_16x16x4_f32(false, a1, false, b1,
                                                     (short)0, acc1, false, false);
    }

    // Diagonal extraction from the 16x16 f32 C/D layout:
    //   lanes 0..15:  VGPR m holds D[M=m][N=lane]     -> diag e<8  in lane e,    vgpr e
    //   lanes 16..31: VGPR m holds D[M=m+8][N=lane-16] -> diag e>=8 in lane e+16, vgpr e-8
    const int j = lane & 7;
    float d = acc0[0] + acc1[0];
    if (j == 1) d = acc0[1] + acc1[1];
    if (j == 2) d = acc0[2] + acc1[2];
    if (j == 3) d = acc0[3] + acc1[3];
    if (j == 4) d = acc0[4] + acc1[4];
    if (j == 5) d = acc0[5] + acc1[5];
    if (j == 6) d = acc0[6] + acc1[6];
    if (j == 7) d = acc0[7] + acc1[7];

    const bool active = (lane < 8) || (lane >= 24);
    if (active) {
        const int e = (lane < 8) ? lane : (lane - 16);
        if (e0 + e < E) {
            out[(long)etype * (long)E + (long)(e0 + e)] = d;
        }
    }
}

extern "C" void kernel_launch(void* const* d_in, const int* in_sizes, int n_in,
                              void* d_out, int out_size, void* d_ws, size_t ws_size,
                              hipStream_t stream) {
    const float* h_user        = (const float*)d_in[0];
    const float* h_item        = (const float*)d_in[1];
    const int*   src_clicks    = (const int*)d_in[2];
    const int*   dst_clicks    = (const int*)d_in[3];
    const int*   src_clickedby = (const int*)d_in[4];
    const int*   dst_clickedby = (const int*)d_in[5];
    const int*   src_follows   = (const int*)d_in[6];
    const int*   dst_follows   = (const int*)d_in[7];
    float*       out           = (float*)d_out;

    const int E        = in_sizes[2];
    const int tilesPer = (E + TILE - 1) / TILE;
    const int blocksX  = (tilesPer + WAVES_PER_BLOCK - 1) / WAVES_PER_BLOCK;

    dim3 grid(blocksX, 3, 1);
    hetero_udotv_wmma_kernel<<<grid, 32 * WAVES_PER_BLOCK, 0, stream>>>(
        h_user, h_item,
        src_clicks, dst_clicks,
        src_clickedby, dst_clickedby,
        src_follows, dst_follows,
        out, E);
}